// EncouragingLoss_22823456211588
// MI455X (gfx1250) — compile-verified
//
#include <hip/hip_runtime.h>

// ---------------------------------------------------------------------------
// EncouragingLoss for MI455X (gfx1250, wave32).
//
// Roofline: 4096 x 32000 x 4B = 512MB streamed once; 23.3 TB/s => ~22us floor.
// ~1.25 exp/element (online LSE) is far below transcendental peak, so the
// kernel is HBM-bound. No matrix contraction exists => no WMMA; the relevant
// CDNA5 path is the async global->LDS copy engine (ASYNCcnt), pipelined
// 8 requests deep per lane via double-buffered 16KB LDS chunks.
// ---------------------------------------------------------------------------

typedef __attribute__((ext_vector_type(4))) float v4f;
typedef __attribute__((ext_vector_type(4))) int   v4i;

#define TPB    256
#define WAVES  (TPB / 32)
#define QI     4                   // async b128 issues per lane per chunk
#define CHUNK  (TPB * 4 * QI)      // floats per LDS chunk (16KB)

__device__ __forceinline__ float fexp2(float x) { return __builtin_amdgcn_exp2f(x); }
__device__ __forceinline__ float flog2(float x) { return __builtin_amdgcn_logf(x); }

#define F_L2E 1.4426950408889634f
#define F_LN2 0.6931471805599453f

// Online log-sum-exp update with a 4-vector (branchless rescale).
__device__ __forceinline__ void upd4(float& m, float& s, v4f v) {
  float mx = fmaxf(fmaxf(v.x, v.y), fmaxf(v.z, v.w));
  float nm = fmaxf(m, mx);
  float b  = nm * F_L2E;
  s = s * fexp2(m * F_L2E - b)
    + fexp2(v.x * F_L2E - b) + fexp2(v.y * F_L2E - b)
    + fexp2(v.z * F_L2E - b) + fexp2(v.w * F_L2E - b);
  m = nm;
}

__device__ __forceinline__ void upd1(float& m, float& s, float v) {
  float nm = fmaxf(m, v);
  float b  = nm * F_L2E;
  s = s * fexp2(m * F_L2E - b) + fexp2(v * F_L2E - b);
  m = nm;
}

// Combine two (max, sumexp) partials.
__device__ __forceinline__ void comb(float& m, float& s, float mo, float so) {
  float nm = fmaxf(m, mo);
  s = s * fexp2((m - nm) * F_L2E) + so * fexp2((mo - nm) * F_L2E);
  m = nm;
}

// CDNA5 async global->LDS path (ASYNCcnt), guarded so the file always compiles.
#if defined(__gfx1250__) && \
    __has_builtin(__builtin_amdgcn_global_load_async_to_lds_b128) && \
    __has_builtin(__builtin_amdgcn_s_wait_asynccnt)
#define EL_ASYNC 1
#endif

#ifdef EL_ASYNC
// Builtin signature (per hipcc diagnostic): param0 is v4i in AS(1) (__device__),
// param1 is the LDS destination in AS(3) (__shared__), then imm offset + cpol.
typedef __attribute__((address_space(1))) v4i* el_gptr;
typedef __attribute__((address_space(3))) v4i* el_lptr;

__device__ __forceinline__ void async_cp16(const float* g, float* l) {
  __builtin_amdgcn_global_load_async_to_lds_b128(
      (el_gptr)(v4i*)(float*)g,
      (el_lptr)(v4i*)l,
      0, 0);
}

// Issue one 16KB chunk: QI b128s per lane, strided 4KB apart.
__device__ __forceinline__ void issue_chunk(const float* gbase, float* lbase, int tid) {
#pragma unroll
  for (int q = 0; q < QI; ++q)
    async_cp16(gbase + q * (TPB * 4) + tid * 4, lbase + q * (TPB * 4) + tid * 4);
}
#endif

__global__ __launch_bounds__(TPB)
void el_row_kernel(const float* __restrict__ inp, const int* __restrict__ tgt,
                   float* __restrict__ rowout, int V) {
  const int row = blockIdx.x;
  const int tid = threadIdx.x;
  const float* rowp = inp + (size_t)row * (size_t)V;

  __shared__ float sh_m[WAVES];
  __shared__ float sh_s[WAVES];

  float m = -1e30f;
  float s = 0.0f;

#ifdef EL_ASYNC
  __shared__ __align__(16) float buf[2][CHUNK];
  if ((V & 3) == 0) {
    const int nfull = V / CHUNK;
    if (nfull > 0) {
      issue_chunk(rowp, &buf[0][0], tid);                 // prologue: chunk 0
      for (int c = 0; c + 1 < nfull; ++c) {
        // Prefetch next chunk into the other buffer, then retire current.
        issue_chunk(rowp + (size_t)(c + 1) * CHUNK, &buf[(c + 1) & 1][0], tid);
        asm volatile("" ::: "memory");
        __builtin_amdgcn_s_wait_asynccnt(QI);             // oldest chunk complete
        asm volatile("" ::: "memory");
        const float* lb = &buf[c & 1][0];
#pragma unroll
        for (int q = 0; q < QI; ++q)
          upd4(m, s, *(const v4f*)(lb + q * (TPB * 4) + tid * 4));
      }
      asm volatile("" ::: "memory");
      __builtin_amdgcn_s_wait_asynccnt(0);
      asm volatile("" ::: "memory");
      const float* lb = &buf[(nfull - 1) & 1][0];
#pragma unroll
      for (int q = 0; q < QI; ++q)
        upd4(m, s, *(const v4f*)(lb + q * (TPB * 4) + tid * 4));
    }
    // vec4 tail past the last full chunk (direct, non-temporal)
    const int base = nfull * CHUNK;
    const v4f* vp = (const v4f*)(rowp + base);
    const int nv4t = (V - base) >> 2;
    for (int i = tid; i < nv4t; i += TPB) upd4(m, s, __builtin_nontemporal_load(vp + i));
  } else {
    for (int j = tid; j < V; j += TPB) upd1(m, s, rowp[j]);
  }
#else
  if ((V & 3) == 0) {
    const v4f* vp = (const v4f*)rowp;
    const int nv4 = V >> 2;
    for (int i = tid; i < nv4; i += TPB) upd4(m, s, __builtin_nontemporal_load(vp + i));
  } else {
    for (int j = tid; j < V; j += TPB) upd1(m, s, rowp[j]);
  }
#endif

  // wave32 butterfly reduce of (m, s)
  for (int off = 16; off > 0; off >>= 1) {
    float mo = __shfl_xor(m, off, 32);
    float so = __shfl_xor(s, off, 32);
    comb(m, s, mo, so);
  }
  const int wave = tid >> 5;
  if ((tid & 31) == 0) { sh_m[wave] = m; sh_s[wave] = s; }
  __syncthreads();

  if (tid == 0) {
    for (int w = 1; w < WAVES; ++w) comb(m, s, sh_m[w], sh_s[w]);
    const int t  = tgt[row];
    const float xt  = rowp[t];
    const float lse = m + flog2(s) * F_LN2;       // natural-log LSE
    const float lp  = xt - lse;                   // lprobs[row, t]
    const float p   = fexp2(lp * F_L2E);          // probs[row, t]
    float bonus;
    if (p > 0.75f) {
      // (p - 0.75)/(0.75 - 1) + log(1 - 0.75)
      bonus = (p - 0.75f) * -4.0f + -1.3862943611198906f;
    } else {
      bonus = flog2(fmaxf(1.0f - p, 1e-5f)) * F_LN2;
    }
    rowout[row] = lp - bonus;                     // gathered adj value
  }
}

// Deterministic final mean: out[0] = -sum(rowout)/N. No atomics -> replays are
// bit-identical and we never rely on the poisoned d_out contents.
__global__ __launch_bounds__(TPB)
void el_reduce_kernel(const float* __restrict__ rowout, float* __restrict__ out, int N) {
  __shared__ float acc[WAVES];
  float s = 0.0f;
  for (int i = threadIdx.x; i < N; i += TPB) s += rowout[i];
  for (int off = 16; off > 0; off >>= 1) s += __shfl_xor(s, off, 32);
  const int wave = threadIdx.x >> 5;
  if ((threadIdx.x & 31) == 0) acc[wave] = s;
  __syncthreads();
  if (threadIdx.x == 0) {
    float t = 0.0f;
    for (int w = 0; w < WAVES; ++w) t += acc[w];
    out[0] = -t / (float)N;
  }
}

extern "C" void kernel_launch(void* const* d_in, const int* in_sizes, int n_in,
                              void* d_out, int out_size, void* d_ws, size_t ws_size,
                              hipStream_t stream) {
  const float* inp = (const float*)d_in[0];
  const int*   tgt = (const int*)d_in[1];   // target indices (per-harness int cast)
  const int N = in_sizes[1];
  const int V = in_sizes[0] / N;

  float* rowout = (float*)d_ws;             // N floats of scratch

  el_row_kernel<<<dim3(N), dim3(TPB), 0, stream>>>(inp, tgt, rowout, V);
  el_reduce_kernel<<<dim3(1), dim3(TPB), 0, stream>>>(rowout, (float*)d_out, N);
}